// NodeEmbeddingModule2_188978561448
// MI455X (gfx1250) — compile-verified
//
#include <hip/hip_runtime.h>
#include <hip/hip_bf16.h>

// Key insight: reference() returns h[14] @ Wo + Bo. Every layer is strictly
// row-wise, so the output depends ONLY on X_v[14] and deg[14]. The whole
// N=250000 dimension is dead code. We compute a chain of 10 GEMVs for one
// row (~2.1 MFLOP over ~4.1 MB of weights, L2-resident, latency-bound),
// launched as 10 small kernels on `stream` (stream order = stage sync).
//
// WMMA mapping (V_WMMA_F32_16X16X4_F32, fp32 in / fp32 acc = exact dtype
// match with the fp32 reference):
//   D(16x16) += A(16x4) x B(4x16)
//   A row 0 = x[k..k+3]      (ISA A layout: lane0={x[k],x[k+1]},
//                             lane16={x[k+2],x[k+3]}, all other lanes 0)
//   B       = W[k..k+3][j0..j0+15]  (VGPR0: rows k,k+2; VGPR1: rows k+1,k+3)
//   y[j0+n] = D[0][n] -> acc[0] in lanes 0..15
// Each wave owns 16 output columns; 4 waves per workgroup.

typedef __attribute__((ext_vector_type(2))) float v2f;
typedef __attribute__((ext_vector_type(8))) float v8f;

#define ACT_NONE 0
#define ACT_RELU 1
#define ACT_TANH_GATE 2

__global__ void __launch_bounds__(128)
gemv_wmma_f32(const float* __restrict__ W,     // (din, dout) row-major
              const float* __restrict__ bias,  // (dout)
              const float* __restrict__ xa,    // first part of input vector
              int lenA,
              const float* __restrict__ xb,    // second part (concat) or null
              float* __restrict__ y,           // (dout)
              int din, int dout, int act,
              const int* __restrict__ deg)
{
    __shared__ float xs[1024];

    const int tid = threadIdx.x;
    for (int i = tid; i < din; i += blockDim.x)
        xs[i] = (i < lenA) ? xa[i] : xb[i - lenA];
    __syncthreads();

    const int lane = tid & 31;
    const int wave = tid >> 5;
    const int j0 = (blockIdx.x * 4 + wave) * 16;
    if (j0 >= dout) return;                 // wave-uniform: EXEC stays all-1s

    const bool lead = (lane & 15) == 0;     // lanes 0 and 16 carry A row 0
    const int  krow = (lane >> 4) << 1;     // 0 for lanes 0-15, 2 for 16-31
    const int  jcol = lane & 15;

    const float* wp = W + (size_t)krow * (size_t)dout + (size_t)(j0 + jcol);
    const size_t step = (size_t)4 * (size_t)dout;

    v8f acc = {};
    for (int k0 = 0; k0 < din; k0 += 4) {
        // A tile: only row M=0 is the vector; all other rows zero.
        const float x0 = xs[k0 + krow];
        const float x1 = xs[k0 + krow + 1];
        v2f a;
        a.x = lead ? x0 : 0.0f;
        a.y = lead ? x1 : 0.0f;
        // B tile: W rows k0+krow (b.x) and k0+krow+1 (b.y), cols j0..j0+15.
        v2f b;
        b.x = wp[0];
        b.y = wp[dout];
        wp += step;
        // (neg_a, A, neg_b, B, c_mod, C, reuse_a, reuse_b)
        acc = __builtin_amdgcn_wmma_f32_16x16x4_f32(
            false, a, false, b, (short)0, acc, false, false);
    }

    if (lane < 16) {                        // D row 0 = lanes 0..15 of acc[0]
        const int j = j0 + jcol;
        float v = acc[0] + bias[j];
        if (act == ACT_RELU) {
            v = fmaxf(v, 0.0f);
        } else if (act == ACT_TANH_GATE) {
            v = tanhf(v);
            if (!(deg[14] > 0)) v = 0.0f;   // jnp.where(has_nbrs, hn, 0)
        }
        y[j] = v;
    }
}

// Input order (setup_inputs dict, lists flattened):
//  0: X_v (250000*512)
//  1..3:  W1[0..2]   4..6:  B1[0..2]
//  7..9:  W2[0..2]  10..12: B2[0..2]
// 13..15: Wl[0..2]  16..18: Bl[0..2]
// 19: Wo (64*64)    20: Bo (64)    21: deg (250000, int32)
extern "C" void kernel_launch(void* const* d_in, const int* in_sizes, int n_in,
                              void* d_out, int out_size, void* d_ws, size_t ws_size,
                              hipStream_t stream)
{
    (void)in_sizes; (void)n_in; (void)out_size; (void)ws_size;

    const float* Xv14 = (const float*)d_in[0] + (size_t)14 * 512;
    const float* W1[3] = {(const float*)d_in[1], (const float*)d_in[2], (const float*)d_in[3]};
    const float* B1[3] = {(const float*)d_in[4], (const float*)d_in[5], (const float*)d_in[6]};
    const float* W2[3] = {(const float*)d_in[7], (const float*)d_in[8], (const float*)d_in[9]};
    const float* B2[3] = {(const float*)d_in[10], (const float*)d_in[11], (const float*)d_in[12]};
    const float* Wl[3] = {(const float*)d_in[13], (const float*)d_in[14], (const float*)d_in[15]};
    const float* Bl[3] = {(const float*)d_in[16], (const float*)d_in[17], (const float*)d_in[18]};
    const float* Wo   = (const float*)d_in[19];
    const float* Bo   = (const float*)d_in[20];
    const int*   deg  = (const int*)d_in[21];

    float* ws = (float*)d_ws;
    // Workspace layout (floats); every region fully written before read.
    float* t1_0 = ws + 0;     // 512
    float* t2g0 = ws + 512;   // 512
    float* h1   = ws + 1024;  // 256
    float* t1_1 = ws + 1280;  // 256
    float* t2g1 = ws + 1536;  // 256
    float* h2   = ws + 1792;  // 128
    float* t1_2 = ws + 1920;  // 128
    float* t2g2 = ws + 2048;  // 128
    float* h3   = ws + 2176;  // 64
    float* out  = (float*)d_out;

    auto launch = [&](const float* W, const float* b,
                      const float* xa, int lenA, const float* xb,
                      float* y, int din, int dout, int act) {
        const int nwg = (dout + 63) / 64;   // 4 waves * 16 cols per WG
        gemv_wmma_f32<<<dim3(nwg), dim3(128), 0, stream>>>(
            W, b, xa, lenA, xb, y, din, dout, act, deg);
    };

    // Layer 0 (d=512, hd=256)
    launch(W1[0], B1[0], Xv14, 512, nullptr, t1_0, 512, 512, ACT_RELU);
    launch(W2[0], B2[0], t1_0, 512, nullptr, t2g0, 512, 512, ACT_TANH_GATE);
    launch(Wl[0], Bl[0], Xv14, 512, t2g0,    h1,  1024, 256, ACT_RELU);
    // Layer 1 (d=256, hd=128)
    launch(W1[1], B1[1], h1,   256, nullptr, t1_1, 256, 256, ACT_RELU);
    launch(W2[1], B2[1], t1_1, 256, nullptr, t2g1, 256, 256, ACT_TANH_GATE);
    launch(Wl[1], Bl[1], h1,   256, t2g1,    h2,   512, 128, ACT_RELU);
    // Layer 2 (d=128, hd=64)
    launch(W1[2], B1[2], h2,   128, nullptr, t1_2, 128, 128, ACT_RELU);
    launch(W2[2], B2[2], t1_2, 128, nullptr, t2g2, 128, 128, ACT_TANH_GATE);
    launch(Wl[2], Bl[2], h2,   128, t2g2,    h3,   256, 64,  ACT_RELU);
    // Output head
    launch(Wo, Bo, h3, 64, nullptr, out, 64, 64, ACT_NONE);
}